// GraphSAGELayer_35914516529155
// MI455X (gfx1250) — compile-verified
//
#include <hip/hip_runtime.h>

typedef __attribute__((ext_vector_type(2))) float v2f;
typedef __attribute__((ext_vector_type(4))) float v4f;
typedef __attribute__((ext_vector_type(8))) float v8f;

#define BM   64      // rows per block
#define DIN  256
#define DH   256
#define KNEI 16

// LDS: s_src[BM*DIN] + s_agg[BM*DIN] + s_w[2*DIN*16]  = 160 KB (<= 320 KB/WGP)
__global__ __launch_bounds__(256, 1)
void sage_fused_kernel(const float* __restrict__ src,
                       const float* __restrict__ nei,
                       const float* __restrict__ W,
                       const float* __restrict__ Wa,
                       float* __restrict__ out) {
    __shared__ float s_src[BM * DIN];
    __shared__ float s_agg[BM * DIN];
    __shared__ float s_w[2 * DIN * 16];

    const int t = threadIdx.x;                 // 0..255
    const int block_row0 = blockIdx.x * BM;

    // ---- Phase A1: stage src strip (coalesced float4) ----
    {
        const v4f* g = (const v4f*)(src + (size_t)block_row0 * DIN);
        v4f* s = (v4f*)s_src;
        const int total = BM * DIN / 4;        // 4096
        for (int i = t; i < total; i += 256) s[i] = g[i];
    }

    // ---- Phase A2: neighbor mean -> s_agg (streams the 1 GB tensor once) ----
    {
        const int total = BM * DIN / 4;        // 4096 float4 chunks
        for (int i = t; i < total; i += 256) {
            const int row = i / (DIN / 4);
            const int c4  = i % (DIN / 4);
            const v4f* g = (const v4f*)(nei + (size_t)(block_row0 + row) * KNEI * DIN) + c4;
            v4f acc = g[0];
#pragma unroll
            for (int k = 1; k < KNEI; ++k) acc += g[(size_t)k * (DIN / 4)];
            acc *= (1.0f / 16.0f);
            ((v4f*)s_agg)[i] = acc;
        }
    }
    __syncthreads();

    // ---- Phase B: WMMA GEMMs ----
    const int lane   = t & 31;
    const int wave   = t >> 5;
    const int half   = lane >> 4;              // 0/1: lane half selects K pair / M group
    const int lm     = lane & 15;              // M (for A) or N (for B/C)
    const int rstrip = wave >> 1;              // 0..3 : 16-row strip
    const int isNei  = wave & 1;               // 0 = self GEMM, 1 = neighbor GEMM

    const float* sA = (isNei ? s_agg : s_src) + (rstrip * 16 + lm) * DIN;
    float* outp = out + (size_t)(block_row0 + rstrip * 16) * (2 * DH) + (isNei ? DH : 0);

    for (int c = 0; c < 16; ++c) {
        __syncthreads();
        // stage W[:, c*16 .. c*16+15] for both matrices into LDS (8192 floats)
        for (int i = t; i < 2 * DIN * 16; i += 256) {
            const int m    = i >> 12;          // 0 = weight, 1 = aggr_weight
            const int r    = (i >> 4) & 255;   // k row
            const int ccol = i & 15;
            const float* Wsrc = m ? Wa : W;
            s_w[i] = Wsrc[r * DH + c * 16 + ccol];
        }
        __syncthreads();

        const float* sW = s_w + isNei * (DIN * 16);
        v8f acc = {};
#pragma unroll 4
        for (int kb = 0; kb < DIN; kb += 4) {
            const int ka = kb + 2 * half;
            v2f a, b;
            // A fragment: A[M=lm][K=ka], A[M=lm][K=ka+1]   (16x4 f32 layout)
            a.x = sA[ka];
            a.y = sA[ka + 1];
            // B fragment: B[K=ka][N=lm], B[K=ka+1][N=lm]   (4x16 f32 layout)
            b.x = sW[ka * 16 + lm];
            b.y = sW[(ka + 1) * 16 + lm];
            acc = __builtin_amdgcn_wmma_f32_16x16x4_f32(
                /*neg_a=*/false, a, /*neg_b=*/false, b,
                /*c_mod=*/(short)0, acc, /*reuse_a=*/false, /*reuse_b=*/false);
        }

        // C/D layout: lane holds D[M = 8*half + j][N = lm]; fuse ReLU + concat
#pragma unroll
        for (int j = 0; j < 8; ++j) {
            const int m = 8 * half + j;
            const int n = c * 16 + lm;
            outp[(size_t)m * (2 * DH) + n] = fmaxf(acc[j], 0.0f);
        }
    }
}

extern "C" void kernel_launch(void* const* d_in, const int* in_sizes, int n_in,
                              void* d_out, int out_size, void* d_ws, size_t ws_size,
                              hipStream_t stream) {
    (void)n_in; (void)out_size; (void)d_ws; (void)ws_size;
    const float* src = (const float*)d_in[0];
    const float* nei = (const float*)d_in[1];
    const float* W   = (const float*)d_in[2];
    const float* Wa  = (const float*)d_in[3];
    float* out       = (float*)d_out;

    const int N = in_sizes[0] / DIN;           // 65536
    dim3 grid(N / BM);                         // 1024 blocks
    sage_fused_kernel<<<grid, 256, 0, stream>>>(src, nei, W, Wa, out);
}